// CheiralityLayer_61813169324727
// MI455X (gfx1250) — compile-verified
//
#include <hip/hip_runtime.h>

typedef __attribute__((ext_vector_type(2))) float v2f;
typedef __attribute__((ext_vector_type(8))) float v8f;

#define NB 32
#define IH 240
#define IW 320
#define NPIX (IH * IW)          // 76800
#define CHUNKS 12               // workgroups per batch per iteration
#define TPB 256                 // 8 waves (wave32)
#define STRIDE (CHUNKS * TPB)   // 3072 = 9*320 + 192
#define STEPS (NPIX / STRIDE)   // 25 (exact)
#define LRC (0.1f / (float)NPIX)
#define EPSV 1e-6f
#define MAXIT 100

// One gradient pass of the cheirality objective. grid = (CHUNKS, NB), block = 256.
// Recomputes gA/gB/nf features from raw inputs each pass (inputs are L2-resident,
// 12 B/pixel — cheaper than caching 28 B/pixel of precomputed features).
__global__ __launch_bounds__(TPB) void cheir_grad_kernel(
    const float* __restrict__ nf, const float* __restrict__ grads,
    const float* __restrict__ pose, float* __restrict__ accum)
{
    const int b = blockIdx.y;
    const int g = blockIdx.x * TPB + (int)threadIdx.x;   // [0, 3072)

    const float* nfb = nf + b * NPIX;
    const float* gxp = grads + b * 2 * NPIX;
    const float* gyp = gxp + NPIX;
    const float* pb  = pose + b * 8;
    const float V0 = pb[0], V1 = pb[1], V2 = pb[2];
    const float O0 = pb[3], O1 = pb[4], O2 = pb[5];

    float s0 = 0.f, s1 = 0.f, s2 = 0.f, s3 = 0.f, s4 = 0.f, s5 = 0.f;
    float xf = (float)(g % IW);
    float yf = (float)(g / IW);
    int idx = g;

#pragma unroll 5
    for (int it = 0; it < STEPS; ++it) {
        float gx = gxp[idx], gy = gyp[idx], nfv = nfb[idx];
        // normalize gradient direction: gd = g / (|g| + eps)
        float gn  = __builtin_amdgcn_sqrtf(__builtin_fmaf(gx, gx, gy * gy));
        float inv = __builtin_amdgcn_rcpf(gn + EPSV);
        gx *= inv; gy *= inv;
        // features
        float gA2 = __builtin_fmaf(gx, xf, gy * yf);
        float xy  = xf * yf;
        float yy1 = __builtin_fmaf(yf, yf, 1.0f);
        float xx1 = __builtin_fmaf(xf, xf, 1.0f);
        float gB0 = __builtin_fmaf(gx, xy, gy * yy1);
        float gB1 = -__builtin_fmaf(gx, xx1, gy * xy);
        float gB2 = __builtin_fmaf(gx, yf, -gy * xf);
        // rho = (gA.V)(nf - gB.Om)
        float a  = __builtin_fmaf(gA2, V2, -__builtin_fmaf(gx, V0, gy * V1));
        float bb = nfv - (__builtin_fmaf(gB0, O0, __builtin_fmaf(gB1, O1, gB2 * O2)));
        float rho = a * bb;
        // dl/drho = -0.5*(1+erf(rho/sqrt2)) - 0.5*rho*sqrt(2/pi)*exp(-rho^2/2)
        // erf via A&S 7.1.26 — shares exp(-z^2) with the derivative term.
        float z  = 0.70710678118f * rho;
        float az = __builtin_fabsf(z);
        float t  = __builtin_amdgcn_rcpf(__builtin_fmaf(0.3275911f, az, 1.0f));
        float E  = __builtin_amdgcn_exp2f(-1.44269504089f * (z * z));
        float poly = t * __builtin_fmaf(t,
                       __builtin_fmaf(t,
                         __builtin_fmaf(t,
                           __builtin_fmaf(t, 1.061405429f, -1.453152027f),
                           1.421413741f),
                         -0.284496736f),
                       0.254829592f);
        float PE   = poly * E;
        float onep = (rho >= 0.0f) ? (2.0f - PE) : PE;  // 1 + erf(z)
        float dl   = __builtin_fmaf(-0.5f, onep, -0.3989422804f * rho * E);
        float w1 = dl * bb;   // weight on gA  (grad wrt V)
        float w2 = -dl * a;   // weight on gB  (grad wrt Om)
        s0 = __builtin_fmaf(w1, -gx, s0);
        s1 = __builtin_fmaf(w1, -gy, s1);
        s2 = __builtin_fmaf(w1, gA2, s2);
        s3 = __builtin_fmaf(w2, gB0, s3);
        s4 = __builtin_fmaf(w2, gB1, s4);
        s5 = __builtin_fmaf(w2, gB2, s5);
        // advance pixel: +3072 = (+9 rows, +192 cols)
        idx += STRIDE;
        xf += 192.0f; yf += 9.0f;
        if (xf >= 320.0f) { xf -= 320.0f; yf += 1.0f; }
    }

    // intra-wave reduction (wave32)
    for (int m = 16; m >= 1; m >>= 1) {
        s0 += __shfl_xor(s0, m, 32);
        s1 += __shfl_xor(s1, m, 32);
        s2 += __shfl_xor(s2, m, 32);
        s3 += __shfl_xor(s3, m, 32);
        s4 += __shfl_xor(s4, m, 32);
        s5 += __shfl_xor(s5, m, 32);
    }

    // per-wave partials -> LDS, padded to 16 components (rows 6..15 zeroed)
    __shared__ float P[8][16];
    const int wave = threadIdx.x >> 5;
    const int lane = threadIdx.x & 31;
    if (lane == 0) {
        P[wave][0] = s0; P[wave][1] = s1; P[wave][2] = s2;
        P[wave][3] = s3; P[wave][4] = s4; P[wave][5] = s5;
        for (int c = 6; c < 16; ++c) P[wave][c] = 0.0f;
    }
    __syncthreads();

    // cross-wave reduction on wave 0 via V_WMMA_F32_16X16X4_F32:
    // A (16x4 f32): M = component, K = wave index; B = ones (layout-invariant)
    // => D[m][n] = sum_k P[wave k][m]; two chained WMMAs cover 8 waves.
    // ISA A layout: lanes 0-15 hold K=0 (v0) / K=1 (v1); lanes 16-31 hold K=2/K=3.
    if (wave == 0) {
        const int m = lane & 15;
        const int half = lane >> 4;
        v2f a0, a1, bones;
        bones.x = 1.0f; bones.y = 1.0f;
        a0.x = P[0 + 2 * half][m];  a0.y = P[1 + 2 * half][m];
        a1.x = P[4 + 2 * half][m];  a1.y = P[5 + 2 * half][m];
        v8f c = {0.f, 0.f, 0.f, 0.f, 0.f, 0.f, 0.f, 0.f};
        c = __builtin_amdgcn_wmma_f32_16x16x4_f32(false, a0, false, bones,
                                                  (short)0, c, false, false);
        c = __builtin_amdgcn_wmma_f32_16x16x4_f32(false, a1, false, bones,
                                                  (short)0, c, false, false);
        // lane 0 column n=0 holds D[m][0] in C components m=0..7
        if (threadIdx.x == 0) {
            atomicAdd(&accum[b * 8 + 0], c[0]);
            atomicAdd(&accum[b * 8 + 1], c[1]);
            atomicAdd(&accum[b * 8 + 2], c[2]);
            atomicAdd(&accum[b * 8 + 3], c[3]);
            atomicAdd(&accum[b * 8 + 4], c[4]);
            atomicAdd(&accum[b * 8 + 5], c[5]);
        }
    }
}

__global__ void cheir_init_kernel(const float* __restrict__ p0,
                                  float* __restrict__ pose,
                                  float* __restrict__ accum)
{
    int i = threadIdx.x;
    if (i < NB * 6) {
        int b = i / 6, c = i % 6;
        pose[b * 8 + c]  = p0[i];
        accum[b * 8 + c] = 0.0f;
    }
}

__global__ void cheir_update_kernel(float* __restrict__ pose,
                                    float* __restrict__ accum)
{
    int i = threadIdx.x;
    if (i < NB * 6) {
        int b = i / 6, c = i % 6;
        int k = b * 8 + c;
        pose[k]  = __builtin_fmaf(-LRC, accum[k], pose[k]);
        accum[k] = 0.0f;   // reset for next iteration
    }
}

__global__ void cheir_out_kernel(const float* __restrict__ pose,
                                 float* __restrict__ out)
{
    int i = threadIdx.x;
    if (i < NB * 6) {
        int b = i / 6, c = i % 6;
        out[i] = pose[b * 8 + c];
    }
}

extern "C" void kernel_launch(void* const* d_in, const int* in_sizes, int n_in,
                              void* d_out, int out_size, void* d_ws, size_t ws_size,
                              hipStream_t stream)
{
    const float* nf    = (const float*)d_in[0];   // (32, 240, 320)
    const float* grads = (const float*)d_in[1];   // (32, 2, 240, 320)
    const float* p0    = (const float*)d_in[2];   // (32, 6)
    float* out   = (float*)d_out;                 // (32, 6)
    float* pose  = (float*)d_ws;                  // 32 * 8 floats (padded)
    float* accum = pose + NB * 8;                 // 32 * 8 floats

    cheir_init_kernel<<<1, 256, 0, stream>>>(p0, pose, accum);
    for (int it = 0; it < MAXIT; ++it) {
        cheir_grad_kernel<<<dim3(CHUNKS, NB), TPB, 0, stream>>>(nf, grads, pose, accum);
        cheir_update_kernel<<<1, 256, 0, stream>>>(pose, accum);
    }
    cheir_out_kernel<<<1, 256, 0, stream>>>(pose, out);
}